// PitchAdaptiveSpectralAnalysis_63634235457572
// MI455X (gfx1250) — compile-verified
//
#include <hip/hip_runtime.h>
#include <hip/hip_bf16.h>
#include <math.h>

// ---------------- problem constants (from reference) ----------------
#define NBATCH       8
#define TSAMP        160000
#define NFRAMES      2001          // T/80 + 1
#define BN           (NBATCH*NFRAMES)   // 16008 frames total
#define MP           16032         // BN padded to multiple of 32
#define FFT          1024
#define L1C          513           // FFT/2 + 1
#define KC           516           // L1 padded to multiple of 4 (GEMM K)
#define NC           544           // L1 padded to multiple of 32 (GEMM N)
#define NSPEC        1056          // 528 cos cols + 528 sin cols
#define RATE_F       15.625f      // 16000/1024
#define FMIN_F       47.012733594515181f   // 3*16000/1021
#define PI_F         3.14159265358979323846f
#define STEP_F       (2.0f*PI_F/1024.0f)

typedef __attribute__((ext_vector_type(2))) float v2f;
typedef __attribute__((ext_vector_type(8))) float v8f;
typedef __attribute__((ext_vector_type(4))) unsigned int u32x4;
typedef __attribute__((ext_vector_type(8))) int i32x8;
typedef __attribute__((ext_vector_type(4))) int i32x4;

__device__ __forceinline__ int iclamp(int v, int lo, int hi) {
    return v < lo ? lo : (v > hi ? hi : v);
}

// ---------------- trig tables ----------------
__global__ void fill_T1(float* __restrict__ T1) {
    int idx = blockIdx.x * blockDim.x + threadIdx.x;
    if (idx >= FFT * NSPEC) return;
    int t = idx / NSPEC, c = idx % NSPEC;
    float v = 0.0f;
    if (c <= 512) {
        int ph = (t * c) & 1023;
        v = cosf((float)ph * STEP_F);
    } else if (c >= 528 && c <= 1040) {
        int k = c - 528;
        int ph = (t * k) & 1023;
        v = -sinf((float)ph * STEP_F);
    }
    T1[idx] = v;
}

// B2[i][j] = w_i * cos(2*pi*i*j/1024), w_i = 1 for i==0 or 512, else 2
__global__ void fill_B2(float* __restrict__ B2) {
    int idx = blockIdx.x * blockDim.x + threadIdx.x;
    if (idx >= KC * NC) return;
    int i = idx / NC, j = idx % NC;
    float v = 0.0f;
    if (i <= 512 && j <= 512) {
        float wgt = (i == 0 || i == 512) ? 1.0f : 2.0f;
        int ph = (i * j) & 1023;
        v = wgt * cosf((float)ph * STEP_F);
    }
    B2[idx] = v;
}

// ---------------- pitch-adaptive windowing ----------------
__global__ void window_kernel(const float* __restrict__ x, const float* __restrict__ f0g,
                              float* __restrict__ w) {
    int f = blockIdx.x;
    int tid = threadIdx.x;
    if (f >= BN) {
        for (int t = tid; t < FFT; t += 256) w[(size_t)f * FFT + t] = 0.0f;
        return;
    }
    int b = f / NFRAMES;
    int n = f % NFRAMES;
    float f0 = f0g[f];
    if (f0 <= FMIN_F) f0 = 500.0f;
    float half = rintf(24000.0f / f0);      // round(1.5*16000/f0)
    int center = n * 80;

    float wv[4], wn[4];
    float sW = 0.0f, sWin = 0.0f;
    #pragma unroll
    for (int i = 0; i < 4; ++i) {
        int t = tid + i * 256;
        int base = t - 512;
        float m = (fabsf((float)base) <= half) ? 1.0f : 0.0f;
        float win = (0.5f * cosf(PI_F * (float)base * f0 * (1.0f / 24000.0f)) + 0.5f) * m;
        int idx = iclamp(center + base, 0, TSAMP - 1);
        float val = x[b * TSAMP + idx] * win;
        wv[i] = val; wn[i] = win;
        sW += val; sWin += win;
    }
    __shared__ float rW[256], rWin[256];
    rW[tid] = sW; rWin[tid] = sWin;
    __syncthreads();
    for (int s = 128; s > 0; s >>= 1) {
        if (tid < s) { rW[tid] += rW[tid + s]; rWin[tid] += rWin[tid + s]; }
        __syncthreads();
    }
    float corr = rW[0] / rWin[0];
    #pragma unroll
    for (int i = 0; i < 4; ++i) {
        int t = tid + i * 256;
        w[(size_t)f * FFT + t] = wv[i] - wn[i] * corr;
    }
}

// ---------------- fp32 WMMA GEMM with TDM-staged B band ----------------
// 4 waves/block; each wave computes one 16x16 tile (block covers 32 rows x 32 cols).
// The block's 32-column band of B ([K x 32], <=128 KB) is staged into LDS once,
// via the Tensor Data Mover (tracked by TENSORcnt), and reused by all waves/all k.
// Fragment layouts per CDNA5 ISA 7.12.2:
//   A 16x4 f32: lane L -> row M = L%16, K-pair = 2*(L/16); VGPR0/1 hold K, K+1
//   B 4x16 f32: lane L -> col N = L%16, VGPR v holds row K = 2*(L/16)+v
//   C/D 16x16:  lane L -> N = L%16, VGPR i holds M = i + 8*(L/16)
__global__ void wmma_gemm(const float* __restrict__ A, const float* __restrict__ Bm,
                          float* __restrict__ C, int Mstore, int Nstore, int K,
                          int lda, int ldb, int ldc, float scale) {
    extern __shared__ float Bs[];          // [K x 32] band of B, row-major
    int lane = threadIdx.x & 31;
    int wave = threadIdx.x >> 5;
    int nband = blockIdx.y * 32;

#if __has_builtin(__builtin_amdgcn_tensor_load_to_lds)
    if (wave == 0) {
        // ---- Tensor DMA descriptor (D#), CDNA5 ISA sec. 8.3/8.4 ----
        unsigned long long gaddr = (unsigned long long)(uintptr_t)(Bm + nband);
        unsigned int ldsOff = (unsigned int)(uintptr_t)(&Bs[0]);
        unsigned int dim0 = (unsigned int)ldb;   // tensor width (elements)
        unsigned int dim1 = (unsigned int)K;     // tensor height

        u32x4 g0;
        g0[0] = 1u;                                              // count=1 (user D#)
        g0[1] = ldsOff;                                          // lds_addr
        g0[2] = (unsigned int)(gaddr & 0xffffffffu);             // global_addr lo
        g0[3] = (unsigned int)((gaddr >> 32) & 0x01ffffffu)      // global_addr hi
              | (2u << 30);                                      // type = 2 ("image")
        i32x8 g1;
        g1[0] = (int)(2u << 16);                                 // data_size = 4 bytes
        g1[1] = (int)((dim0 & 0xffffu) << 16);                   // tensor_dim0 lo16
        g1[2] = (int)((dim0 >> 16) | ((dim1 & 0xffffu) << 16));  // dim0 hi | dim1 lo
        g1[3] = (int)((dim1 >> 16) | (32u << 16));               // dim1 hi | tile_dim0=32
        g1[4] = (int)(K & 0xffff);                               // tile_dim1=K, tile_dim2=0
        g1[5] = (int)ldb;                                        // tensor_dim0_stride lo32
        g1[6] = 0;                                               // stride0 hi | stride1 lo
        g1[7] = 0;
        i32x4 gz = {0, 0, 0, 0};
#if defined(__clang_major__) && __clang_major__ >= 23
        i32x8 gz8 = {0, 0, 0, 0, 0, 0, 0, 0};
        __builtin_amdgcn_tensor_load_to_lds(g0, g1, gz, gz, gz8, 0);
#else
        __builtin_amdgcn_tensor_load_to_lds(g0, g1, gz, gz, 0);
#endif
        __builtin_amdgcn_s_wait_tensorcnt(0);                    // s_wait_tensorcnt 0
    }
#else
    // fallback: cooperative staged copy
    for (int i = threadIdx.x; i < K * 32; i += 128)
        Bs[i] = Bm[(size_t)(i >> 5) * ldb + nband + (i & 31)];
#endif
    __syncthreads();

    int m0 = (blockIdx.x * 2 + (wave & 1)) * 16;
    int nt = (wave >> 1) * 16;                 // 0 or 16 within the band
    int g = lane >> 4;                         // K-pair group (0 or 1)
    int r = lane & 15;

    const float* arow = A + (size_t)(m0 + r) * lda + 2 * g;      // 8B-aligned (lda even)
    const float* brow = Bs + 2 * g * 32 + nt + r;

    v8f acc = {};
    for (int k0 = 0; k0 < K; k0 += 4) {
        if ((k0 & 63) == 0 && k0 + 64 < K)
            __builtin_prefetch(arow + k0 + 64, 0, 1);            // global_prefetch
        v2f a, b;
        const float2 av = *reinterpret_cast<const float2*>(arow + k0);
        a[0] = av.x; a[1] = av.y;
        b[0] = brow[k0 * 32];
        b[1] = brow[(k0 + 1) * 32];
        // D = A*B + C, full fp32 (V_WMMA_F32_16X16X4_F32)
        acc = __builtin_amdgcn_wmma_f32_16x16x4_f32(
            /*neg_a=*/false, a, /*neg_b=*/false, b,
            /*c_mod=*/(short)0, acc, /*reuse_a=*/false, /*reuse_b=*/false);
    }
    int nn = nband + nt + r;
    #pragma unroll
    for (int i = 0; i < 8; ++i) {
        int mm = m0 + i + 8 * g;
        if (mm < Mstore && nn < Nstore)
            C[(size_t)mm * ldc + nn] = acc[i] * scale;
    }
}

// ---------------- power spectrum -> DC correction -> linear smoothing -> log ----------------
__global__ void spectral_kernel(const float* __restrict__ spec, const float* __restrict__ f0g,
                                float* __restrict__ lgps) {
    int f = blockIdx.x;
    int tid = threadIdx.x;
    if (f >= BN) {
        for (int k = tid; k < KC; k += 256) lgps[(size_t)f * KC + k] = 0.0f;
        return;
    }
    __shared__ float ps[L1C];
    __shared__ float ps2[L1C];
    __shared__ float ext[641];
    __shared__ float Cs[642];

    float f0 = f0g[f];
    if (f0 <= FMIN_F) f0 = 500.0f;

    for (int k = tid; k < L1C; k += 256) {
        float c = spec[(size_t)f * NSPEC + k];
        float s = spec[(size_t)f * NSPEC + 528 + k];
        ps[k] = c * c + s * s;
    }
    __syncthreads();
    for (int k = tid; k < L1C; k += 256) {
        float pos = f0 * (1.0f / RATE_F) - (float)k;
        int lo = iclamp((int)floorf(pos), 0, L1C - 2);
        float fr = pos - (float)lo;
        float rep = ps[lo] * (1.0f - fr) + ps[lo + 1] * fr;
        ps2[k] = ps[k] + (((float)k * RATE_F < f0) ? rep : 0.0f);
    }
    __syncthreads();
    for (int i = tid; i < 641; i += 256) {
        int e = i - 64;
        int j = e < 0 ? -e : e;
        if (j > 512) j = 1024 - j;
        ext[i] = ps2[j];
    }
    __syncthreads();
    if (tid == 0) {
        Cs[0] = 0.0f;
        float acc = 0.0f;
        for (int i = 0; i < 641; ++i) { acc += ext[i]; Cs[i + 1] = acc * RATE_F; }
    }
    __syncthreads();
    float width = f0 * (2.0f / 3.0f);
    float hwb = width / (2.0f * RATE_F);
    for (int k = tid; k < KC; k += 256) {
        if (k >= L1C) { lgps[(size_t)f * KC + k] = 0.0f; continue; }
        float qh = (float)k + hwb + 64.5f;
        float ql = (float)k - hwb + 64.5f;
        int loh = iclamp((int)floorf(qh), 0, 640);
        float frh = qh - (float)loh;
        float vh = Cs[loh] * (1.0f - frh) + Cs[loh + 1] * frh;
        int lol = iclamp((int)floorf(ql), 0, 640);
        float frl = ql - (float)lol;
        float vl = Cs[lol] * (1.0f - frl) + Cs[lol + 1] * frl;
        float val = (vh - vl) / width + 1.1920929e-7f;
        lgps[(size_t)f * KC + k] = logf(fmaxf(val, 1e-30f));
    }
}

// ---------------- cepstral liftering ----------------
__global__ void lifter_kernel(const float* __restrict__ cep, const float* __restrict__ f0g,
                              float* __restrict__ lift) {
    int idx = blockIdx.x * blockDim.x + threadIdx.x;
    if (idx >= MP * KC) return;
    int f = idx / KC;
    int n = idx % KC;
    float v = 0.0f;
    if (f < BN && n < L1C) {
        float f0 = f0g[f];
        if (f0 <= FMIN_F) f0 = 500.0f;
        float z = f0 * (float)n * (1.0f / 16000.0f);
        float sl = 1.0f;
        if (n > 0) { float pz = PI_F * z; sl = sinf(pz) / pz; }
        float cl = 1.3f - 0.3f * cosf(2.0f * PI_F * z);   // 1 - 2*Q1 + 2*Q1*cos, Q1=-0.15
        v = cep[(size_t)f * NC + n] * sl * cl;
    }
    lift[idx] = v;
}

// ---------------- host launch ----------------
extern "C" void kernel_launch(void* const* d_in, const int* in_sizes, int n_in,
                              void* d_out, int out_size, void* d_ws, size_t ws_size,
                              hipStream_t stream) {
    const float* x  = (const float*)d_in[0];   // (8, 160000)
    const float* f0 = (const float*)d_in[1];   // (8, 2001)
    float* out = (float*)d_out;                // (8, 2001, 513)

    float* ws = (float*)d_ws;
    size_t off = 0;
    float* Wbuf = ws + off;  off += (size_t)MP * FFT;      // windowed frames   [MP x 1024]
    float* T1   = ws + off;  off += (size_t)FFT * NSPEC;   // DFT cos/-sin      [1024 x 1056]
    float* Spec = ws + off;  off += (size_t)MP * NSPEC;    // cos/sin sums      [MP x 1056]
    float* Lgps = ws + off;  off += (size_t)MP * KC;       // log smoothed ps   [MP x 516]
    float* B2   = ws + off;  off += (size_t)KC * NC;       // weighted cosine   [516 x 544]
    float* Cep  = ws + off;  off += (size_t)MP * NC;       // cepstrum          [MP x 544]
    float* Lift = ws + off;  off += (size_t)MP * KC;       // lifted cepstrum   [MP x 516]

    fill_T1<<<(FFT * NSPEC + 255) / 256, 256, 0, stream>>>(T1);
    fill_B2<<<(KC * NC + 255) / 256, 256, 0, stream>>>(B2);

    window_kernel<<<MP, 256, 0, stream>>>(x, f0, Wbuf);

    // rfft as GEMM: Spec = Wbuf x T1 ; B band (1024x32 = 128 KB) TDM-staged in LDS
    wmma_gemm<<<dim3(MP / 32, NSPEC / 32), 128, (size_t)FFT * 32 * sizeof(float), stream>>>(
        Wbuf, T1, Spec, MP, NSPEC, FFT, FFT, NSPEC, NSPEC, 1.0f);

    spectral_kernel<<<MP, 256, 0, stream>>>(Spec, f0, Lgps);

    // irfft(log ps): Cep = (1/1024) * Lgps x B2 ; band = 516x32 = 66 KB
    wmma_gemm<<<dim3(MP / 32, NC / 32), 128, (size_t)KC * 32 * sizeof(float), stream>>>(
        Lgps, B2, Cep, MP, NC, KC, KC, NC, NC, 1.0f / 1024.0f);

    lifter_kernel<<<((size_t)MP * KC + 255) / 256, 256, 0, stream>>>(Cep, f0, Lift);

    // hfft(lifted) -> d_out
    wmma_gemm<<<dim3(MP / 32, NC / 32), 128, (size_t)KC * 32 * sizeof(float), stream>>>(
        Lift, B2, out, BN, L1C, KC, KC, NC, L1C, 1.0f);

    (void)in_sizes; (void)n_in; (void)out_size; (void)ws_size;
}